// GCNModel_90331752169512
// MI455X (gfx1250) — compile-verified
//
#include <hip/hip_runtime.h>
#include <math.h>

// GAT (2 heads, 128 ch) + edge MLP for MI455X (gfx1250, wave32).
// h = x@W.T done with V_WMMA_F32_16X16X4_F32 (exact fp32 matrix pipe).
// Scatter softmax via native u32-encoded atomic max + relaxed f32 atomic add.

#define IN_DIM  256
#define OUT_DIM 256     // HEADS*HID
#define NEG_ATT 0.2f
#define NEG_ACT 0.01f

typedef __attribute__((ext_vector_type(2))) float v2f;
typedef __attribute__((ext_vector_type(8))) float v8f;

__device__ __forceinline__ float leaky(float x, float s) { return x > 0.f ? x : s * x; }

// Monotone float<->uint mapping so atomicMax(u32) == float max (handles negatives).
__device__ __forceinline__ unsigned enc_f32(float f) {
  unsigned u = __float_as_uint(f);
  return (u & 0x80000000u) ? ~u : (u | 0x80000000u);
}
__device__ __forceinline__ float dec_f32(unsigned u) {
  return (u & 0x80000000u) ? __uint_as_float(u ^ 0x80000000u) : __uint_as_float(~u);
}

__device__ __forceinline__ void fatomic_add(float* p, float v) {
  __hip_atomic_fetch_add(p, v, __ATOMIC_RELAXED, __HIP_MEMORY_SCOPE_AGENT);
}

// ---------------------------------------------------------------------------
// Kernel 1: h = x @ W.T   x:[N,256], W:[256,256] (row-major), h:[N,256]
// Block = 256 thr = 8 waves; block owns a 16-row M tile; wave w owns columns
// [32w,32w+32) as two 16x16 WMMA accumulators.  K marched in steps of 4.
// ---------------------------------------------------------------------------
__global__ void gat_gemm_kernel(const float* __restrict__ x,
                                const float* __restrict__ W,
                                float* __restrict__ h, int nNodes) {
  int m0 = blockIdx.x * 16;
#if defined(__gfx1250__) && __has_builtin(__builtin_amdgcn_wmma_f32_16x16x4_f32)
  int lane = threadIdx.x & 31;
  int wave = threadIdx.x >> 5;          // 0..7
  int l16  = lane & 15;
  int half = lane >> 4;
  int n0a  = wave * 32;
  int n0b  = n0a + 16;
  int rowA = m0 + l16;
  if (rowA >= nNodes) rowA = nNodes - 1;          // clamp: keep EXEC all-ones
  const float* ap  = x + (size_t)rowA * IN_DIM + 2 * half;          // A[m][k]
  const float* bpa = W + (size_t)(n0a + l16) * IN_DIM + 2 * half;   // B[k][n]=W[n][k]
  const float* bpb = W + (size_t)(n0b + l16) * IN_DIM + 2 * half;
  v8f acc0 = {}; v8f acc1 = {};
#pragma unroll 4
  for (int k0 = 0; k0 < IN_DIM; k0 += 4) {
    v2f a  = *(const v2f*)(ap + k0);
    v2f b0 = *(const v2f*)(bpa + k0);
    v2f b1 = *(const v2f*)(bpb + k0);
    acc0 = __builtin_amdgcn_wmma_f32_16x16x4_f32(false, a, false, b0,
                                                 (short)0, acc0, false, false);
    acc1 = __builtin_amdgcn_wmma_f32_16x16x4_f32(false, a, false, b1,
                                                 (short)0, acc1, false, false);
  }
#pragma unroll
  for (int r = 0; r < 8; ++r) {
    int row = m0 + r + 8 * half;                  // C layout: VGPR r, lane half
    if (row < nNodes) {
      h[(size_t)row * OUT_DIM + n0a + l16] = acc0[r];
      h[(size_t)row * OUT_DIM + n0b + l16] = acc1[r];
    }
  }
#else
  // Scalar fallback (also the host-pass sema body).
  int col = threadIdx.x;
  for (int r = 0; r < 16; ++r) {
    int row = m0 + r;
    if (row < nNodes) {
      float s = 0.f;
      for (int k = 0; k < IN_DIM; ++k)
        s += x[(size_t)row * IN_DIM + k] * W[(size_t)col * IN_DIM + k];
      h[(size_t)row * OUT_DIM + col] = s;
    }
  }
#endif
}

// ---------------------------------------------------------------------------
// Kernel 2: a_src[n,h] = <h[n,h,:],att_src[h,:]>, same for a_dst. Wave/node.
// Lane covers 8 contiguous channels; lanes 0-15 = head 0, 16-31 = head 1.
// ---------------------------------------------------------------------------
__global__ void att_proj_kernel(const float* __restrict__ h,
                                const float* __restrict__ att_src,
                                const float* __restrict__ att_dst,
                                float* __restrict__ a_src,
                                float* __restrict__ a_dst, int nNodes) {
  int wave = (int)((blockIdx.x * blockDim.x + threadIdx.x) >> 5);
  int lane = threadIdx.x & 31;
  if (wave >= nNodes) return;
  int head  = lane >> 4;
  int cbase = (lane & 15) * 8;
  const float* hp = h + (size_t)wave * OUT_DIM + lane * 8;
  float ps = 0.f, pd = 0.f;
#pragma unroll
  for (int i = 0; i < 8; ++i) {
    float v = hp[i];
    ps += v * att_src[head * 128 + cbase + i];
    pd += v * att_dst[head * 128 + cbase + i];
  }
  for (int off = 1; off < 16; off <<= 1) {        // reduce within 16-lane group
    ps += __shfl_xor(ps, off, 32);
    pd += __shfl_xor(pd, off, 32);
  }
  if ((lane & 15) == 0) {
    a_src[wave * 2 + head] = ps;
    a_dst[wave * 2 + head] = pd;
  }
}

// ---------------------------------------------------------------------------
// Kernel 3: zero the accumulators (must run every launch; harness poisons ws).
// ---------------------------------------------------------------------------
__global__ void init_kernel(float* __restrict__ out, float* __restrict__ z,
                            unsigned* __restrict__ menc, int nNodes) {
  size_t i = (size_t)blockIdx.x * blockDim.x + threadIdx.x;
  if (i < (size_t)nNodes * OUT_DIM) out[i] = 0.f;
  if (i < (size_t)nNodes * 2) { z[i] = 0.f; menc[i] = 0u; }  // 0u == encoded -max
}

// ---------------------------------------------------------------------------
// Kernel 4: segment max of e = leaky(a_src[s]+a_dst[d], 0.2) over dst.
// Edges i<E come from edge_index, i>=E are self-loops.
// ---------------------------------------------------------------------------
__global__ void edge_max_kernel(const int* __restrict__ ei,
                                const float* __restrict__ a_src,
                                const float* __restrict__ a_dst,
                                unsigned* __restrict__ menc, int nEdges, int nNodes) {
  int i = blockIdx.x * blockDim.x + threadIdx.x;
  if (i >= nEdges + nNodes) return;
  int s, d;
  if (i < nEdges) { s = ei[i]; d = ei[nEdges + i]; } else { s = d = i - nEdges; }
#pragma unroll
  for (int hd = 0; hd < 2; ++hd) {
    float e = leaky(a_src[s * 2 + hd] + a_dst[d * 2 + hd], NEG_ATT);
    atomicMax(&menc[d * 2 + hd], enc_f32(e));     // native GLOBAL_ATOMIC_MAX_U32
  }
}

// Kernel 5: z[d,h] += exp(e - m[d,h])
__global__ void edge_sum_kernel(const int* __restrict__ ei,
                                const float* __restrict__ a_src,
                                const float* __restrict__ a_dst,
                                const unsigned* __restrict__ menc,
                                float* __restrict__ z, int nEdges, int nNodes) {
  int i = blockIdx.x * blockDim.x + threadIdx.x;
  if (i >= nEdges + nNodes) return;
  int s, d;
  if (i < nEdges) { s = ei[i]; d = ei[nEdges + i]; } else { s = d = i - nEdges; }
#pragma unroll
  for (int hd = 0; hd < 2; ++hd) {
    float e = leaky(a_src[s * 2 + hd] + a_dst[d * 2 + hd], NEG_ATT);
    fatomic_add(&z[d * 2 + hd], expf(e - dec_f32(menc[d * 2 + hd])));
  }
}

// ---------------------------------------------------------------------------
// Kernel 6: out[d,:] += alpha * h[s,:].  Wave per edge; lane owns 8 channels
// (32 B contiguous per lane -> coalesced 1 KB gather per wave).
// ---------------------------------------------------------------------------
__global__ void edge_agg_kernel(const int* __restrict__ ei,
                                const float* __restrict__ a_src,
                                const float* __restrict__ a_dst,
                                const unsigned* __restrict__ menc,
                                const float* __restrict__ z,
                                const float* __restrict__ h,
                                float* __restrict__ out, int nEdges, int nNodes) {
  int wave = blockIdx.x * (blockDim.x >> 5) + (threadIdx.x >> 5);
  int lane = threadIdx.x & 31;
  if (wave >= nEdges + nNodes) return;
  int s, d;
  if (wave < nEdges) { s = ei[wave]; d = ei[nEdges + wave]; } else { s = d = wave - nEdges; }
  float alpha = 0.f;
  if (lane < 2) {                                  // lane == head
    int hd = lane;
    float e  = leaky(a_src[s * 2 + hd] + a_dst[d * 2 + hd], NEG_ATT);
    float ex = expf(e - dec_f32(menc[d * 2 + hd]));
    alpha = ex / z[d * 2 + hd];
  }
  float a0 = __shfl(alpha, 0, 32);
  float a1 = __shfl(alpha, 1, 32);
  float ah = (lane < 16) ? a0 : a1;                // lane's head
  const float* hp = h + (size_t)s * OUT_DIM + lane * 8;
  float*       op = out + (size_t)d * OUT_DIM + lane * 8;
#pragma unroll
  for (int i = 0; i < 8; ++i) fatomic_add(&op[i], ah * hp[i]);
}

// Kernel 7: out = leaky(out + bias, 0.01) in place
__global__ void finalize_kernel(float* __restrict__ out,
                                const float* __restrict__ bias, size_t total) {
  size_t i = (size_t)blockIdx.x * blockDim.x + threadIdx.x;
  if (i >= total) return;
  out[i] = leaky(out[i] + bias[i & (OUT_DIM - 1)], NEG_ACT);
}

// ---------------------------------------------------------------------------
// Kernel 8: y[e,:] = fc1_w @ concat(h2[src],h2[dst]) + fc1_b.  Wave per edge;
// lanes 0-15 cover the src row, 16-31 the dst row (16 floats each, float4 IO).
// fc1_w (3x512 = 6 KB) staged in LDS once per block.
// ---------------------------------------------------------------------------
__global__ void edge_mlp_kernel(const float* __restrict__ h2,
                                const int* __restrict__ ei,
                                const float* __restrict__ fc1_w,
                                const float* __restrict__ fc1_b,
                                float* __restrict__ y, int nEdges) {
  __shared__ float wl[3 * 512];
  for (int i = threadIdx.x; i < 3 * 512; i += blockDim.x) wl[i] = fc1_w[i];
  __syncthreads();
  int wave = blockIdx.x * (blockDim.x >> 5) + (threadIdx.x >> 5);
  int lane = threadIdx.x & 31;
  if (wave >= nEdges) return;
  int s = ei[wave], d = ei[nEdges + wave];
  int k = lane * 16;                               // position in 512-wide rep
  const float* rp = (k < 256) ? (h2 + (size_t)s * OUT_DIM + k)
                              : (h2 + (size_t)d * OUT_DIM + (k - 256));
  float p0 = 0.f, p1 = 0.f, p2 = 0.f;
#pragma unroll
  for (int i = 0; i < 16; i += 4) {
    float4 r = *(const float4*)(rp + i);
    const float* w0 = wl + 0 * 512 + k + i;
    const float* w1 = wl + 1 * 512 + k + i;
    const float* w2 = wl + 2 * 512 + k + i;
    p0 += r.x * w0[0] + r.y * w0[1] + r.z * w0[2] + r.w * w0[3];
    p1 += r.x * w1[0] + r.y * w1[1] + r.z * w1[2] + r.w * w1[3];
    p2 += r.x * w2[0] + r.y * w2[1] + r.z * w2[2] + r.w * w2[3];
  }
  for (int off = 16; off >= 1; off >>= 1) {
    p0 += __shfl_xor(p0, off, 32);
    p1 += __shfl_xor(p1, off, 32);
    p2 += __shfl_xor(p2, off, 32);
  }
  if (lane == 0) {
    y[(size_t)wave * 3 + 0] = p0 + fc1_b[0];
    y[(size_t)wave * 3 + 1] = p1 + fc1_b[1];
    y[(size_t)wave * 3 + 2] = p2 + fc1_b[2];
  }
}

extern "C" void kernel_launch(void* const* d_in, const int* in_sizes, int n_in,
                              void* d_out, int out_size, void* d_ws, size_t ws_size,
                              hipStream_t stream) {
  const float* x       = (const float*)d_in[0];
  const int*   ei      = (const int*)d_in[1];   // [2,E]: row0=src, row1=dst
  const float* W       = (const float*)d_in[2];
  const float* att_src = (const float*)d_in[3];
  const float* att_dst = (const float*)d_in[4];
  const float* bias    = (const float*)d_in[5];
  const float* fc1_w   = (const float*)d_in[6];
  const float* fc1_b   = (const float*)d_in[7];

  const int N = in_sizes[0] / IN_DIM;
  const int E = in_sizes[1] / 2;

  // Workspace layout (~52 MB): h | out | a_src | a_dst | z | m_enc
  float*    h     = (float*)d_ws;
  float*    out   = h + (size_t)N * OUT_DIM;
  float*    a_src = out + (size_t)N * OUT_DIM;
  float*    a_dst = a_src + (size_t)N * 2;
  float*    z     = a_dst + (size_t)N * 2;
  unsigned* menc  = (unsigned*)(z + (size_t)N * 2);
  float*    y     = (float*)d_out;

  const int EN  = E + N;
  const size_t tot = (size_t)N * OUT_DIM;

  gat_gemm_kernel<<<(N + 15) / 16, 256, 0, stream>>>(x, W, h, N);
  att_proj_kernel<<<(N + 7) / 8, 256, 0, stream>>>(h, att_src, att_dst, a_src, a_dst, N);
  init_kernel<<<(unsigned)((tot + 255) / 256), 256, 0, stream>>>(out, z, menc, N);
  edge_max_kernel<<<(EN + 255) / 256, 256, 0, stream>>>(ei, a_src, a_dst, menc, E, N);
  edge_sum_kernel<<<(EN + 255) / 256, 256, 0, stream>>>(ei, a_src, a_dst, menc, z, E, N);
  edge_agg_kernel<<<(EN + 7) / 8, 256, 0, stream>>>(ei, a_src, a_dst, menc, z, h, out, E, N);
  finalize_kernel<<<(unsigned)((tot + 255) / 256), 256, 0, stream>>>(out, bias, tot);
  edge_mlp_kernel<<<(E + 7) / 8, 256, 0, stream>>>(out, ei, fc1_w, fc1_b, y, E);
}